// Rudy_24816321037014
// MI455X (gfx1250) — compile-verified
//
#include <hip/hip_runtime.h>
#include <hip/hip_bf16.h>

// RUDY congestion map for MI455X (gfx1250, wave32).
// Structure: zero maps -> per-net bbox + WMMA-batched 4x4 overlap outer
// products -> f32 global atomics into L2-resident h/v maps -> combine.

#define NBX 512
#define NBY 512
#define NBINS (NBX * NBY)

__constant__ const float kXL = 0.0f, kYL = 0.0f;
#define BSX 2.0f
#define BSY 2.0f
#define FTINY 1.17549435e-38f   // smallest normal f32 (jnp.finfo(f32).tiny)

typedef __attribute__((ext_vector_type(2))) float v2f;
typedef __attribute__((ext_vector_type(8))) float v8f;

__global__ void rudy_zero(float* __restrict__ ws) {
    int i = blockIdx.x * blockDim.x + threadIdx.x;
    if (i < 2 * NBINS) ws[i] = 0.0f;
}

__global__ __launch_bounds__(256) void rudy_scatter(
    const float* __restrict__ pin_pos,      // [2*n_total] (x then y)
    const float* __restrict__ net_weights,  // [num_nets]
    const int*   __restrict__ netpin_start, // [num_nets+1]
    const int*   __restrict__ flat_netpin,  // [n_flat]
    float* __restrict__ hmap,               // [NBINS]
    float* __restrict__ vmap,               // [NBINS]
    int num_nets, int n_total)
{
    // 12 floats per net slot: ox[4], oy[4], hs, vs, ixl, iyl
    __shared__ float sm[256 * 12];

    const int tid = threadIdx.x;
    const int net = blockIdx.x * 256 + tid;

    float ox[4] = {0.f, 0.f, 0.f, 0.f};
    float oy[4] = {0.f, 0.f, 0.f, 0.f};
    float hs = 0.f, vs = 0.f;
    int ixl = 0, iyl = 0;

    if (net < num_nets) {
        const int s = netpin_start[net];
        const int e = netpin_start[net + 1];
        float xmn = 3.4e38f, xmx = -3.4e38f, ymn = 3.4e38f, ymx = -3.4e38f;
        for (int p = s; p < e; ++p) {
            const int idx = flat_netpin[p];
            const float x = pin_pos[idx];
            const float y = pin_pos[n_total + idx];
            xmn = fminf(xmn, x); xmx = fmaxf(xmx, x);
            ymn = fminf(ymn, y); ymx = fmaxf(ymx, y);
        }
        if (e > s) {
            const float w = net_weights[net];
            ixl = min(max((int)((xmn - kXL) / BSX), 0), NBX - 1);
            const int ixh = min((int)((xmx - kXL) / BSX) + 1, NBX);
            iyl = min(max((int)((ymn - kYL) / BSY), 0), NBY - 1);
            const int iyh = min((int)((ymx - kYL) / BSY) + 1, NBY);
            hs = w / (ymx - ymn + FTINY);   // weight / dy  (H utilization)
            vs = w / (xmx - xmn + FTINY);   // weight / dx  (V utilization)
#pragma unroll
            for (int i = 0; i < 4; ++i) {
                const int bx = ixl + i, by = iyl + i;
                const float blx = kXL + (float)bx * BSX;
                const float bly = kYL + (float)by * BSY;
                const float o1 = fminf(xmx, blx + BSX) - fmaxf(xmn, blx);
                const float o2 = fminf(ymx, bly + BSY) - fmaxf(ymn, bly);
                ox[i] = (bx < ixh) ? fmaxf(o1, 0.f) : 0.f;
                oy[i] = (by < iyh) ? fmaxf(o2, 0.f) : 0.f;
            }
        }
    }

    // Stage this wave's 32 nets into LDS for WMMA packing.
    float* slot = &sm[tid * 12];
#pragma unroll
    for (int i = 0; i < 4; ++i) { slot[i] = ox[i]; slot[4 + i] = oy[i]; }
    slot[8] = hs; slot[9] = vs;
    slot[10] = __int_as_float(ixl);
    slot[11] = __int_as_float(iyl);
    __syncthreads();

    // WMMA batching: V_WMMA_F32_16X16X4_F32 computes 4 nets' 4x4 overlap
    // outer products as block-diagonal 4x4 tiles of the 16x16 D matrix.
    //   A[4q+i][k] = (k==q) ? ox_q[i] : 0     (16x4)
    //   B[k][4q+j] = (k==q) ? oy_q[j] : 0     (4x16)
    //   => D[4q+i][4q'+j] = (q==q') * ox_q[i]*oy_q[j]
    // A layout (2 VGPRs): lanes 0-15 hold rows M=lane with K=0 (v0), K=1 (v1);
    //                     lanes 16-31 hold rows M=lane-16 with K=2, K=3.
    // B layout (2 VGPRs): v0: lanes 0-15 row K=0, lanes 16-31 row K=1;
    //                     v1: lanes 0-15 row K=2, lanes 16-31 row K=3.
    // D layout: lanes 0-15 v[r] = (M=r, N=lane); lanes 16-31 v[r] = (M=r+8, N=lane-16).
    const int lane  = tid & 31;
    const int wbase = (tid >> 5) * 32;      // wave's first slot
    const int half  = lane >> 4;            // 0: lanes 0-15, 1: lanes 16-31
    const int m     = lane & 15;            // row index (A) == col index (B/D)
    const int q     = m >> 2;               // net-within-group this lane touches
    const int ij    = m & 3;                // i for A rows, j for B cols
    const int kA0   = half ? 2 : 0, kA1 = half ? 3 : 1;
    const int kB0   = half ? 1 : 0, kB1 = half ? 3 : 2;
    // Diagonal-block holders: lanes 0-7 (nets q=0,1) and 24-31 (nets q=2,3).
    const bool active = half ? (q >= 2) : (q <= 1);
    const bool hi     = (q & 1) != 0;       // net's rows live in D[4..7]

    for (int G = 0; G < 8; ++G) {           // 8 groups of 4 nets per wave
        const float* sl = &sm[(wbase + 4 * G + q) * 12];
        const float a0 = (q == kA0) ? sl[ij]     : 0.f;
        const float a1 = (q == kA1) ? sl[ij]     : 0.f;
        const float b0 = (q == kB0) ? sl[4 + ij] : 0.f;
        const float b1 = (q == kB1) ? sl[4 + ij] : 0.f;
        v2f A = {a0, a1};
        v2f B = {b0, b1};
        v8f C = {0.f, 0.f, 0.f, 0.f, 0.f, 0.f, 0.f, 0.f};
        // Uniform control flow: EXEC is all-ones here (WMMA requirement).
        v8f D = __builtin_amdgcn_wmma_f32_16x16x4_f32(
            false, A, false, B, (short)0, C, false, false);

        if (active) {
            // Constant-index selects (v_cndmask), no dynamic vector indexing
            // -> avoids the compiler spilling D through LDS on the scatter path.
            float dsel[4];
            dsel[0] = hi ? D[4] : D[0];
            dsel[1] = hi ? D[5] : D[1];
            dsel[2] = hi ? D[6] : D[2];
            dsel[3] = hi ? D[7] : D[3];

            const float shs  = sl[8];
            const float svs  = sl[9];
            const int   bixl = __float_as_int(sl[10]);
            const int   col  = __float_as_int(sl[11]) + ij;  // iyl + j
#pragma unroll
            for (int r = 0; r < 4; ++r) {
                const float d = dsel[r];
                if (d != 0.f) {                 // zero iff invalid/empty bin
                    const int bin = (bixl + r) * NBY + col;
                    atomicAdd(hmap + bin, d * shs);
                    atomicAdd(vmap + bin, d * svs);
                }
            }
        }
    }
}

__global__ void rudy_combine(const float* __restrict__ h,
                             const float* __restrict__ v,
                             float* __restrict__ out)
{
    int i = blockIdx.x * blockDim.x + threadIdx.x;
    if (i < NBINS) {
        const float sc = 1.0f / (BSX * BSY * 1.5f);  // 1/(bin_area*UNIT_CAP)
        out[i] = fmaxf(fabsf(h[i]) * sc, fabsf(v[i]) * sc);
    }
}

extern "C" void kernel_launch(void* const* d_in, const int* in_sizes, int n_in,
                              void* d_out, int out_size, void* d_ws, size_t ws_size,
                              hipStream_t stream)
{
    const float* pin_pos      = (const float*)d_in[0];
    const float* net_weights  = (const float*)d_in[1];
    const int*   netpin_start = (const int*)d_in[2];
    const int*   flat_netpin  = (const int*)d_in[3];
    const int num_nets = in_sizes[1];
    const int n_total  = in_sizes[0] / 2;

    float* hmap = (float*)d_ws;
    float* vmap = hmap + NBINS;

    rudy_zero<<<(2 * NBINS + 255) / 256, 256, 0, stream>>>(hmap);
    rudy_scatter<<<(num_nets + 255) / 256, 256, 0, stream>>>(
        pin_pos, net_weights, netpin_start, flat_netpin,
        hmap, vmap, num_nets, n_total);
    rudy_combine<<<(NBINS + 255) / 256, 256, 0, stream>>>(
        hmap, vmap, (float*)d_out);
}